// SparseMaxPool_19232863551513
// MI455X (gfx1250) — compile-verified
//
#include <hip/hip_runtime.h>
#include <stdint.h>

// CDNA5 / gfx1250. wave32.
typedef float        v4f  __attribute__((ext_vector_type(4)));
typedef unsigned int u32x4 __attribute__((ext_vector_type(4)));
typedef int          i32x4 __attribute__((ext_vector_type(4)));
typedef int          i32x8 __attribute__((ext_vector_type(8)));

#define NN 64

// Sparse mask of valid (i, j) pairs, d = j - i (derived from COUNTS=(15,8,8)):
//   level 0: d in [0,14], any i
//   level 1: d in {16,18,...,30}, i even
//   level 2: d in {34,38,...,62}, d = 34 + 4t, i % 4 == 0
__device__ __forceinline__ bool pair_on(int i, int d) {
    if (d <= 14) return true;
    if (d >= 16 && d <= 30) return ((d & 1) == 0) && ((i & 1) == 0);
    if (d >= 34) return (((d - 34) & 3) == 0) && ((i & 3) == 0);
    return false;
}

__global__ void __launch_bounds__(256) sparse_maxpool_kernel(
        const float* __restrict__ x, float* __restrict__ out, int rows) {
    __shared__ __align__(16) float s_x[NN];          // one (b,c) row, DMA'd by TDM
    __shared__ __align__(16) float s_tile[NN * NN];  // 16 KB output tile

    const int r   = blockIdx.x;    // r = b*C + c
    const int tid = threadIdx.x;

    // 1) Zero-fill the output tile in LDS (overlaps with the TDM DMA below).
    {
        v4f z = {0.f, 0.f, 0.f, 0.f};
        v4f* t4 = (v4f*)s_tile;
#pragma unroll
        for (int c = 0; c < 4; ++c) t4[tid + 256 * c] = z;
    }

    // 2) Wave 0 issues a Tensor-Data-Mover DMA: 64 x f32 row, global -> LDS.
    //    TDM ignores EXEC; whole waves with EXEC==0 branch over this block, so the
    //    descriptor is issued exactly once per workgroup. Tracked by TENSORcnt.
    if (tid == 0) {
        uint64_t ga      = (uint64_t)(const void*)(x + (size_t)r * NN);
        uint32_t lds_off = (uint32_t)(size_t)(&s_x[0]);  // low 32 bits of generic addr = LDS byte offset

        u32x4 g0;
        g0[0] = 1u;                                   // count = 1 (valid user descriptor)
        g0[1] = lds_off;                              // lds_addr (bytes)
        g0[2] = (uint32_t)(ga & 0xffffffffu);         // global_addr[31:0]
        g0[3] = (uint32_t)((ga >> 32) & 0x1ffffffu)   // global_addr[56:32]
              | (2u << 30);                           // D#.type = 2 ("image")

        i32x8 g1;
        g1[0] = 0x20000;        // workgroup_mask=0 (not in cluster), data_size=2 -> 4 bytes
        g1[1] = (NN << 16);     // tensor_dim0 = 64  (bits 79:48, low half)
        g1[2] = (1  << 16);     // tensor_dim1 = 1
        g1[3] = (NN << 16);     // tile_dim0   = 64  (bits 127:112)
        g1[4] = 1;              // tile_dim1   = 1
        g1[5] = NN;             // tensor_dim0_stride = 64 (elements)
        g1[6] = 0;
        g1[7] = 0;

        i32x4 gz4 = {0, 0, 0, 0};            // groups 2/3 unused (<=2D tensor)
        i32x8 gz8 = {0, 0, 0, 0, 0, 0, 0, 0};
        __builtin_amdgcn_tensor_load_to_lds(g0, g1, gz4, gz4, gz8, 0);
        __builtin_amdgcn_s_wait_tensorcnt(0);         // s_wait_tensorcnt 0
    }
    __syncthreads();   // publishes DMA'd s_x and the zeroed tile to all 8 waves

    // 3) Threads 0..63: running segment max along row i; scatter to masked (i,j).
    //    value(i,j) = max(x[i..j]) exactly matches the pooling-pyramid recurrence.
    if (tid < NN) {
        const int i = tid;
        float m = s_x[i];
        s_tile[i * NN + i] = m;                       // d = 0
        for (int j = i + 1; j < NN; ++j) {
            m = fmaxf(m, s_x[j]);
            if (pair_on(i, j - i)) s_tile[i * NN + j] = m;
        }
    }
    __syncthreads();

    // 4) Stream the 16 KB tile out: coalesced 128-bit non-temporal stores
    //    (256 MB total write stream; NT avoids rinsing it through L2).
    {
        const v4f* t4 = (const v4f*)s_tile;
        v4f* o4 = (v4f*)(out + (size_t)r * (NN * NN));
#pragma unroll
        for (int c = 0; c < 4; ++c)
            __builtin_nontemporal_store(t4[tid + 256 * c], o4 + tid + 256 * c);
    }

    // 5) Block 0 also emits mask2d as float 0/1, appended after x2d.
    if (r == 0) {
        float* mo = out + (size_t)rows * (NN * NN);
        for (int e = tid; e < NN * NN; e += 256) {
            int i = e / NN, j = e % NN;
            mo[e] = (j >= i && pair_on(i, j - i)) ? 1.0f : 0.0f;
        }
    }
}

extern "C" void kernel_launch(void* const* d_in, const int* in_sizes, int n_in,
                              void* d_out, int out_size, void* d_ws, size_t ws_size,
                              hipStream_t stream) {
    const float* x = (const float*)d_in[0];   // [B, C, 64] f32, B*C*64 elements
    float* out = (float*)d_out;               // [B*C*64*64] x2d + [64*64] mask (floats)
    const int rows = in_sizes[0] / NN;        // B*C = 16384
    sparse_maxpool_kernel<<<dim3(rows), dim3(256), 0, stream>>>(x, out, rows);
}